// SpecialSpmm_28475633173126
// MI455X (gfx1250) — compile-verified
//
#include <hip/hip_runtime.h>
#include <stdint.h>

// Batched sparse-COO @ dense scatter:
//   out[g, row_e, :] += values[g, e] * b[g, col_e, :]
// B=4, E=800000, N=50000, D=64.
//
// Memory-bound: ~1.7 GB of L2-resident traffic (b and out each 51 MB < 192 MB L2).
// Strategy: wave32-cooperative edges. 32 edges of metadata are async-DMA'd into
// a per-wave LDS double buffer (CDNA5 global_load_async_to_lds_b32 / ASYNCcnt),
// then broadcast lane-by-lane with v_readlane; all 32 lanes gather one 256B
// b-row (float2/lane) and scatter with fire-and-forget global_atomic_add_f32.

namespace {

constexpr int       kB     = 4;
constexpr int       kE     = 800000;
constexpr int       kN     = 50000;
constexpr int       kD     = 64;
constexpr int       kWave  = 32;          // gfx1250 is wave32-only
constexpr int       kWavesPerBlock = 8;   // 256 threads
constexpr int       kBlock = kWave * kWavesPerBlock;
constexpr int       kChunk = kWave;       // edges staged per wave iteration
constexpr int       kChunks = kE / kChunk;   // 25000 (exact)
constexpr int       kBlocksX = 256;          // persistent blocks per batch
constexpr long long kOutElems = (long long)kB * kN * kD;  // 12.8M floats

// Low 32 bits of a generic pointer to __shared__ data == LDS byte address.
__device__ __forceinline__ uint32_t lds_addr32(const void* p) {
  return (uint32_t)(uintptr_t)p;
}

// CDNA5 async DMA: each lane copies 4 bytes from the global address in vaddr to the LDS address in vdst, tracked by ASYNCcnt.
__device__ __forceinline__ void async_load_b32(uint32_t lds_byte, const void* gptr) {
  asm volatile("global_load_async_to_lds_b32 %0, %1, off"
               :
               : "v"(lds_byte), "v"(gptr)
               : "memory");
}

template <int N>
__device__ __forceinline__ void wait_asynccnt() {
  asm volatile("s_wait_asynccnt %0" : : "i"(N) : "memory");
}

__global__ void zero_out_kernel(float4* __restrict__ out) {
  const long long i = (long long)blockIdx.x * blockDim.x + threadIdx.x;
  out[i] = make_float4(0.f, 0.f, 0.f, 0.f);
}

__global__ __launch_bounds__(kBlock) void spmm_scatter_kernel(
    const int*   __restrict__ indices,  // [B, 2, E] (int32 per harness convention)
    const float* __restrict__ values,   // [B, E]
    const float* __restrict__ bmat,     // [B, N, D]
    float*       __restrict__ out)      // [B, N, D]
{
  __shared__ int   s_row[kWavesPerBlock][2][kWave];
  __shared__ int   s_col[kWavesPerBlock][2][kWave];
  __shared__ float s_val[kWavesPerBlock][2][kWave];

  const int lane = (int)(threadIdx.x & (kWave - 1));
  const int wid  = (int)(threadIdx.x >> 5);   // wave32
  const int g    = (int)blockIdx.y;           // batch

  const int*    rowp = indices + (long long)g * 2 * kE;
  const int*    colp = rowp + kE;
  const float*  valp = values + (long long)g * kE;
  const float2* bg   = (const float2*)(bmat + (long long)g * kN * kD);
  float*        og   = out + (long long)g * kN * kD;

  const int nwaves = kBlocksX * kWavesPerBlock;
  int chunk = (int)blockIdx.x * kWavesPerBlock + wid;
  if (chunk >= kChunks) return;

  auto issue = [&](int buf, int ch) {
    const int e = ch * kChunk + lane;  // coalesced: lane-contiguous metadata
    async_load_b32(lds_addr32(&s_row[wid][buf][lane]), rowp + e);
    async_load_b32(lds_addr32(&s_col[wid][buf][lane]), colp + e);
    async_load_b32(lds_addr32(&s_val[wid][buf][lane]), valp + e);
  };

  issue(0, chunk);
  int buf  = 0;
  int next = chunk + nwaves;

  for (;;) {
    const bool have_next = (next < kChunks);
    if (have_next) {
      issue(buf ^ 1, next);     // prefetch next chunk's metadata
      wait_asynccnt<3>();       // in-order: current buffer's 3 loads are done
    } else {
      wait_asynccnt<0>();
    }

    const int   r = s_row[wid][buf][lane];
    const int   c = s_col[wid][buf][lane];
    const float v = s_val[wid][buf][lane];

#pragma unroll 8
    for (int j = 0; j < kWave; ++j) {
      // Broadcast edge j's metadata to SGPRs (uniform across the wave).
      const int   rj = __builtin_amdgcn_readlane(r, j);
      const int   cj = __builtin_amdgcn_readlane(c, j);
      const float vj = __int_as_float(__builtin_amdgcn_readlane(__float_as_int(v), j));

      // 32 lanes x float2 = one coalesced 256B row gather of b[g, cj, :].
      const float2 src = bg[(long long)cj * (kD / 2) + lane];
      float* dst = og + (long long)rj * kD + lane * 2;

      // Fire-and-forget f32 atomics (no return -> STOREcnt, resolved in L2).
      unsafeAtomicAdd(dst + 0, vj * src.x);
      unsafeAtomicAdd(dst + 1, vj * src.y);
    }

    if (!have_next) break;
    chunk = next;
    next += nwaves;
    buf ^= 1;
  }
}

}  // namespace

extern "C" void kernel_launch(void* const* d_in, const int* in_sizes, int n_in,
                              void* d_out, int out_size, void* d_ws, size_t ws_size,
                              hipStream_t stream) {
  (void)in_sizes; (void)n_in; (void)out_size; (void)d_ws; (void)ws_size;

  // setup_inputs order: indices [B,2,E] int, values [B,E] f32, n (scalar), b [B,N,D] f32
  const int*   indices = (const int*)d_in[0];
  const float* values  = (const float*)d_in[1];
  const float* bmat    = (const float*)d_in[3];
  float*       out     = (float*)d_out;

  // Zero the accumulator (harness poisons d_out; atomics accumulate).
  // 12.8M floats = 3.2M float4 = 12500 blocks x 256 threads, exact.
  zero_out_kernel<<<dim3((unsigned)(kOutElems / 4 / kBlock)), dim3(kBlock), 0, stream>>>(
      (float4*)out);

  spmm_scatter_kernel<<<dim3(kBlocksX, kB), dim3(kBlock), 0, stream>>>(
      indices, values, bmat, out);
}